// LAGCN_77129022701602
// MI455X (gfx1250) — compile-verified
//
#include <hip/hip_runtime.h>
#include <hip/hip_bf16.h>

// ---------------------------------------------------------------------------
// LA-GCN forward for MI455X (gfx1250): fp32 WMMA GEMMs + TDM tile staging +
// fp32 global-atomic segment sums. Bandwidth-bound workload (~8 GB traffic).
// ---------------------------------------------------------------------------

typedef float  v2f  __attribute__((ext_vector_type(2)));
typedef float  v8f  __attribute__((ext_vector_type(8)));
typedef unsigned int u32x4 __attribute__((ext_vector_type(4)));
typedef int    i32x4 __attribute__((ext_vector_type(4)));
typedef int    i32x8 __attribute__((ext_vector_type(8)));

#if __has_builtin(__builtin_amdgcn_tensor_load_to_lds)
#define HAVE_TDM 1
#if __has_include(<hip/amd_detail/amd_gfx1250_TDM.h>)
#define CDNA5_TDM_6ARG 1   // amdgpu-toolchain (clang-23): 6-arg builtin
#endif
#else
#define HAVE_TDM 0
#endif

__device__ __forceinline__ void atomic_add_f(float* p, float v) {
  unsafeAtomicAdd(p, v);   // global_atomic_add_f32
}

#if HAVE_TDM
// Issue one TDM 2D tile load: tile_h rows of tile_w fp32 elements, global row
// stride row_stride (elements), into LDS at byte offset lds_off. Rows beyond
// tensor_h read as zero (OOB fill). Optional LDS padding: +1 DWORD every 32
// DWORDs stored (pad_interval code 4 => 32 DWORDs, pad_amount code 0 => 1).
__device__ __forceinline__ void tdm_load_2d(unsigned lds_off, const void* gptr,
                                            unsigned tile_w, unsigned tile_h,
                                            unsigned tensor_w, unsigned tensor_h,
                                            unsigned long long row_stride,
                                            bool pad) {
  unsigned long long ga = (unsigned long long)gptr;
  u32x4 g0;
  g0[0] = 1u;                                    // count=1, user descriptor
  g0[1] = lds_off;                               // lds_addr (bytes)
  g0[2] = (unsigned)ga;                          // global_addr[31:0]
  g0[3] = ((unsigned)(ga >> 32) & 0x01FFFFFFu)   // global_addr[56:32]
        | (2u << 30);                            // type=2 ("image")
  unsigned d0 = (2u << 16);                      // data_size=2 -> 4 bytes
  if (pad) d0 |= (1u << 20) | (4u << 22);        // pad_enable, interval=32 DW, amount=1 DW
  i32x8 g1;
  g1[0] = (int)d0;
  g1[1] = (int)((tensor_w & 0xFFFFu) << 16);                     // tensor_dim0[15:0]
  g1[2] = (int)((tensor_w >> 16) | ((tensor_h & 0xFFFFu) << 16));// dim0 hi | dim1 lo
  g1[3] = (int)((tensor_h >> 16) | ((tile_w & 0xFFFFu) << 16));  // dim1 hi | tile_dim0
  g1[4] = (int)(tile_h & 0xFFFFu);                               // tile_dim1 (tile_dim2=0)
  g1[5] = (int)(row_stride & 0xFFFFFFFFull);                     // tensor_dim0_stride lo
  g1[6] = (int)((row_stride >> 32) & 0xFFFFull);                 // stride0 hi (stride1=0)
  g1[7] = 0;
  i32x4 z4 = {0, 0, 0, 0};
#if defined(CDNA5_TDM_6ARG)
  i32x8 z8 = {0, 0, 0, 0, 0, 0, 0, 0};
  __builtin_amdgcn_tensor_load_to_lds(g0, g1, z4, z4, z8, 0);
#else
  __builtin_amdgcn_tensor_load_to_lds(g0, g1, z4, z4, 0);
#endif
}
#endif // HAVE_TDM

// ---------------------------------------------------------------------------
// WMMA fp32 GEMM: C[M x NC] = A[M x Kd] @ B[Kd x NC]; ldA = Kd, ldC = NC.
// NC is compile-time so all B-fragment LDS addressing is constant-folded
// (unpredicated ds_load_2addr), and partial-subtile handling is static.
// Block = 256 threads (8 waves). Block tile: 128 rows x (NSUB*16) cols.
// Wave w computes rows [m0 + 16w, m0 + 16w + 16) x all NSUB col subtiles.
// K staged in KT=32 chunks through LDS (TDM when available).
// ---------------------------------------------------------------------------
#define KT 32
#define BM 128

template <int NSUB, int NC>
__global__ __launch_bounds__(256) void gemm_wmma_f32(
    const float* __restrict__ A, const float* __restrict__ B,
    float* __restrict__ C, int M, int Kd) {
  constexpr int B_ELEMS = KT * 16 * NSUB;        // B tile capacity (floats)
  __shared__ float smem[B_ELEMS + BM * 33 + 64]; // B tile | A tile (33-f rows)
  float* Bs = smem;
  float* As = smem + B_ELEMS;

  const int tid  = threadIdx.x;
  const int wave = tid >> 5;
  const int lane = tid & 31;
  const int half = lane >> 4;   // K/M-half select per ISA layouts
  const int mn   = lane & 15;   // A row within strip / B-D column
  const int m0   = blockIdx.x * BM;

  v8f acc[NSUB] = {};

  const int rem = M - m0;
  const unsigned tenH = rem < BM ? (unsigned)rem : (unsigned)BM;

  for (int k0 = 0; k0 < Kd; k0 += KT) {
    __syncthreads();                       // previous tile fully consumed
#if HAVE_TDM
    if (tid == 0) {
      // B tile: KT x NC, packed rows in LDS at offset 0
      tdm_load_2d(0u, B + (size_t)k0 * (size_t)NC,
                  (unsigned)NC, KT, (unsigned)NC, KT,
                  (unsigned long long)NC, false);
      // A tile: BM x KT, LDS rows padded to 33 DWORDs (conflict-free)
      tdm_load_2d((unsigned)(B_ELEMS * 4), A + (size_t)m0 * (size_t)Kd + k0,
                  KT, BM, KT, tenH, (unsigned long long)Kd, true);
    }
    __builtin_amdgcn_s_wait_tensorcnt(0);
#else
    for (int idx = tid; idx < KT * NC; idx += 256) {
      int r = idx / NC, c2 = idx % NC;
      Bs[idx] = B[(size_t)(k0 + r) * (size_t)NC + c2];
    }
    for (int idx = tid; idx < BM * KT; idx += 256) {
      int r = idx / KT, c2 = idx % KT;
      float v = (m0 + r < M) ? A[(size_t)(m0 + r) * (size_t)Kd + (k0 + c2)] : 0.0f;
      As[r * 33 + c2] = v;
    }
#endif
    __syncthreads();

#pragma unroll
    for (int kk = 0; kk < KT; kk += 4) {
      // A fragment: lane(half,mn) holds A[m][half*2 + j]  (ISA 16x4 f32 layout)
      v2f a;
      a.x = As[(wave * 16 + mn) * 33 + kk + half * 2 + 0];
      a.y = As[(wave * 16 + mn) * 33 + kk + half * 2 + 1];
#pragma unroll
      for (int nt = 0; nt < NSUB; ++nt) {
        const int col = nt * 16 + mn;
        // Unpredicated LDS reads: for a partial last subtile (NC % 16 != 0)
        // the extra columns read in-bounds LDS junk that only feeds
        // accumulator lanes whose outputs are never stored.
        v2f b;
        b.x = Bs[(kk + half * 2 + 0) * NC + col];
        b.y = Bs[(kk + half * 2 + 1) * NC + col];
        acc[nt] = __builtin_amdgcn_wmma_f32_16x16x4_f32(
            false, a, false, b, (short)0, acc[nt], false, false);
      }
    }
  }

  // D layout: VGPR r holds M = r (lanes 0-15) / M = 8+r (lanes 16-31), N = mn.
#pragma unroll
  for (int nt = 0; nt < NSUB; ++nt) {
    const int col = nt * 16 + mn;
    const bool full = (nt + 1) * 16 <= NC;       // compile-time per subtile
    if (!full && col >= NC) continue;
#pragma unroll
    for (int r = 0; r < 8; ++r) {
      const int row = m0 + wave * 16 + half * 8 + r;
      if (row < M) C[(size_t)row * (size_t)NC + col] = acc[nt][r];
    }
  }
}

// ---------------------------------------------------------------------------
// Degree / normalization
// ---------------------------------------------------------------------------
__global__ void deg_init(float* deg, int n) {
  int i = blockIdx.x * blockDim.x + threadIdx.x;
  if (i < n) deg[i] = 1.0f;                      // self-loop
}
__global__ void deg_count(const int* __restrict__ ei, float* deg, int E) {
  int e = blockIdx.x * blockDim.x + threadIdx.x;
  if (e < E) atomic_add_f(&deg[ei[(size_t)E + e]], 1.0f);
}
__global__ void deg_rsqrt(float* deg, int n) {
  int i = blockIdx.x * blockDim.x + threadIdx.x;
  if (i < n) deg[i] = rsqrtf(deg[i]);
}

// ---------------------------------------------------------------------------
// Layer 1 aggregation: hcat[:, k*128 : (k+1)*128]
// ---------------------------------------------------------------------------
__global__ void hcat_self(const float* __restrict__ dinv,
                          const float* __restrict__ xwb,
                          float* __restrict__ hcat, int N, int k) {
  long long idx = (long long)blockIdx.x * blockDim.x + threadIdx.x;
  if (idx >= (long long)N * 128) return;
  int i = (int)(idx >> 7), f = (int)(idx & 127);
  float di = dinv[i];
  hcat[(size_t)i * 512 + (size_t)k * 128 + f] = di * di * xwb[idx];
}

__global__ void agg_edges(const int* __restrict__ ei,
                          const float* __restrict__ dinv,
                          const float* __restrict__ xwb,
                          float* __restrict__ hcat, int E, int k) {
  long long gid = (long long)blockIdx.x * blockDim.x + threadIdx.x;
  long long e = gid >> 7;
  int f = (int)(gid & 127);
  if (e >= E) return;
  int s = ei[e];
  int d = ei[(long long)E + e];
  if (f == 0) {
    long long e2 = e + 1024;
    if (e2 < E) __builtin_prefetch(&xwb[(size_t)ei[e2] * 128], 0, 0);
  }
  float c = dinv[s] * dinv[d];
  atomic_add_f(&hcat[(size_t)d * 512 + (size_t)k * 128 + f],
               c * xwb[(size_t)s * 128 + f]);
}

__global__ void bias_relu(float* __restrict__ hcat,
                          const float* __restrict__ b1, long long total) {
  long long idx = (long long)blockIdx.x * blockDim.x + threadIdx.x;
  if (idx >= total) return;
  int j = (int)(idx & 511);                      // b1 flat [K*128]
  float v = hcat[idx] + b1[j];
  hcat[idx] = v > 0.0f ? v : 0.0f;
}

// ---------------------------------------------------------------------------
// Layer 2 aggregation: out[N, 40]
// ---------------------------------------------------------------------------
__global__ void out_self(const float* __restrict__ dinv,
                         const float* __restrict__ xw2,
                         float* __restrict__ out, int N) {
  long long idx = (long long)blockIdx.x * blockDim.x + threadIdx.x;
  if (idx >= (long long)N * 40) return;
  int i = (int)(idx / 40);
  float di = dinv[i];
  out[idx] = di * di * xw2[idx];
}

__global__ void out_edges(const int* __restrict__ ei,
                          const float* __restrict__ dinv,
                          const float* __restrict__ xw2,
                          float* __restrict__ out, int E) {
  long long gid = (long long)blockIdx.x * blockDim.x + threadIdx.x;
  long long e = gid / 40;
  int c = (int)(gid % 40);
  if (e >= E) return;
  int s = ei[e];
  int d = ei[(long long)E + e];
  float cf = dinv[s] * dinv[d];
  atomic_add_f(&out[(size_t)d * 40 + c], cf * xw2[(size_t)s * 40 + c]);
}

__global__ void out_bias(float* __restrict__ out, const float* __restrict__ b2,
                         int N) {
  long long idx = (long long)blockIdx.x * blockDim.x + threadIdx.x;
  if (idx >= (long long)N * 40) return;
  out[idx] += b2[(int)(idx % 40)];
}

// ---------------------------------------------------------------------------
static inline unsigned div_up_u(long long a, long long b) {
  return (unsigned)((a + b - 1) / b);
}

extern "C" void kernel_launch(void* const* d_in, const int* in_sizes, int n_in,
                              void* d_out, int out_size, void* d_ws, size_t ws_size,
                              hipStream_t stream) {
  const float* x_list = (const float*)d_in[0];   // [K, N, 256]
  const int*   ei     = (const int*)d_in[1];     // [2, E]
  const float* W1     = (const float*)d_in[2];   // [K, 256, 128]
  const float* b1     = (const float*)d_in[3];   // [K, 128]
  const float* W2     = (const float*)d_in[4];   // [512, 40]
  const float* b2     = (const float*)d_in[5];   // [40]
  float* out = (float*)d_out;                    // [N, 40]

  const int DIN = 256, DH = 128, NCLS = 40, K = 4;
  const int E = in_sizes[1] / 2;
  const int N = in_sizes[0] / (K * DIN);

  // Workspace layout (floats): dinv | xw_branch | hcat | xw2  (~136 MB)
  float* ws   = (float*)d_ws;
  float* dinv = ws;                              // N (padded to 50048)
  float* xwb  = dinv + 50048;                    // N*128
  float* hcat = xwb + (size_t)N * DH;            // N*512
  float* xw2  = hcat + (size_t)N * (K * DH);     // N*40

  // --- normalization coefficients ---
  deg_init <<<div_up_u(N, 256), 256, 0, stream>>>(dinv, N);
  deg_count<<<div_up_u(E, 256), 256, 0, stream>>>(ei, dinv, E);
  deg_rsqrt<<<div_up_u(N, 256), 256, 0, stream>>>(dinv, N);

  const dim3 gemm_grid(div_up_u(N, BM));

  // --- layer 1, branch-serial to bound scratch ---
  for (int k = 0; k < K; ++k) {
    gemm_wmma_f32<8, 128><<<gemm_grid, 256, 0, stream>>>(
        x_list + (size_t)k * N * DIN, W1 + (size_t)k * DIN * DH,
        xwb, N, DIN);
    hcat_self<<<div_up_u((long long)N * DH, 256), 256, 0, stream>>>(
        dinv, xwb, hcat, N, k);
    agg_edges<<<div_up_u((long long)E * DH, 256), 256, 0, stream>>>(
        ei, dinv, xwb, hcat, E, k);
  }
  bias_relu<<<div_up_u((long long)N * (K * DH), 256), 256, 0, stream>>>(
      hcat, b1, (long long)N * (K * DH));

  // --- layer 2 ---
  gemm_wmma_f32<3, 40><<<gemm_grid, 256, 0, stream>>>(
      hcat, W2, xw2, N, K * DH);
  out_self <<<div_up_u((long long)N * NCLS, 256), 256, 0, stream>>>(dinv, xw2, out, N);
  out_edges<<<div_up_u((long long)E * NCLS, 256), 256, 0, stream>>>(ei, dinv, xw2, out, E);
  out_bias <<<div_up_u((long long)N * NCLS, 256), 256, 0, stream>>>(out, b2, N);
}